// ColorSaliency_77773267796185
// MI455X (gfx1250) — compile-verified
//
#include <hip/hip_runtime.h>

// ---------------------------------------------------------------------------
// ColorSaliency on gfx1250 (MI455X):
//   sal = mean_c | frame - boxmean5x5(frame) | ;  out = sal / (max(sal)+1e-6)
//
// Separable 5x5 SAME box filter as banded GEMMs on the matrix engine
// (V_WMMA_F32_16X16X4_F32, exact f32, K accumulated in chunks of 4).
// Each wave produces a 16x32 output tile from a 20x36 zero-padded patch:
//   stage 1:  T(16x48) = Wv(16x20) x P(20x48)        (3 N-tiles x 5 K-chunks)
//   stage 2:  Out(16x32) = T x Wh                    (2 out-tiles x 5 K-chunks)
// Band matrices Wv[m][r] = Wh[r][m] = (m <= r <= m+4) have IDENTICAL per-lane
// register images in A-layout and B-layout, so one set of 5 v2f weights
// serves both stages.
//
// Roofline: ~192 MiB frame read (x1.41 halo, mostly L2-absorbed: L2 = 192 MB)
// + 64 MiB sal write + 128 MiB normalize r/w  =>  ~17-20 us at 23.3 TB/s.
// Global max via 1024 distributed atomicMax slots (<=32 same-address RMWs
// each) + a tiny reduce kernel, avoiding ~32K serialized L2 atomics on one
// address which would otherwise dominate the runtime.
// ---------------------------------------------------------------------------

typedef __attribute__((ext_vector_type(2))) float v2f;
typedef __attribute__((ext_vector_type(8))) float v8f;

#define IMG_H 1024
#define IMG_W 1024
#define NCH   3
#define NB    16
#define NSLOT 1024          // distributed-max slots in d_ws (uint bits)
#define PSTR  48            // P row stride (floats), cols 36..47 zero
#define TSTR  49            // T row stride; gcd(49,64)=1 -> conflict-free A reads

// One wave (32 lanes) per 16x32 output tile, all 3 channels.
__global__ void __launch_bounds__(32)
sal5x5_wmma_kernel(const float* __restrict__ frame,
                   float* __restrict__ out,
                   unsigned int* __restrict__ slots)
{
    __shared__ float P[20 * PSTR];   // padded input patch
    __shared__ float T[16 * TSTR];   // vertical box sums (cols 0..47)

    const int lane = threadIdx.x;        // 0..31
    const int hf   = lane >> 4;          // lane-half
    const int lcol = lane & 15;          // M(A) / N(B) / N(C,D) index
    const int k0   = 2 * hf;             // K-local of v2f component 0
    const int k1   = 2 * hf + 1;

    const int x0 = blockIdx.x << 5;      // 32-wide tiles
    const int y0 = blockIdx.y << 4;
    const int b  = blockIdx.z;

    // Banded weights: comp j = ((4k + kloc_j) in [lcol, lcol+4]).
    // Same image for Wv chunks (A-layout) and Wh chunks (B-layout).
    v2f wband[5];
#pragma unroll
    for (int k = 0; k < 5; ++k) {
        const int r0 = 4 * k + k0, r1 = 4 * k + k1;
        wband[k].x = (r0 >= lcol && r0 <= lcol + 4) ? 1.f : 0.f;
        wband[k].y = (r1 >= lcol && r1 <= lcol + 4) ? 1.f : 0.f;
    }

    // Patch columns handled by this lane in the global->LDS load.
    const int  gx1 = x0 - 2 + lane;                 // cols 0..31
    const int  gx2 = x0 + 30 + lane;                // cols 32..35 (lane<4)
    const bool cv1 = (gx1 >= 0) && (gx1 < IMG_W);
    const bool cv2 = (lane < 4) && (gx2 < IMG_W);

    float sal[2][8];
#pragma unroll
    for (int t = 0; t < 2; ++t)
#pragma unroll
        for (int i = 0; i < 8; ++i) sal[t][i] = 0.f;

    for (int ch = 0; ch < NCH; ++ch) {
        const float* src = frame + (size_t)(b * NCH + ch) * (IMG_H * IMG_W);

        // ---- load 20x36 patch, zero padding to 20x48 ----
#pragma unroll
        for (int r = 0; r < 20; ++r) {
            const int  gy = y0 - 2 + r;
            const bool rv = (gy >= 0) && (gy < IMG_H);
            float v1 = 0.f, v2 = 0.f;
            if (rv && cv1) v1 = src[gy * IMG_W + gx1];
            if (rv && cv2) v2 = src[gy * IMG_W + gx2];
            P[r * PSTR + lane] = v1;
            if (lane < 16) P[r * PSTR + 32 + lane] = v2;   // cols 32..47
        }
        __syncthreads();

        // ---- stage 1: T = Wv(16x20) x P(20x48) ----
        v8f acc0 = {}, acc1 = {}, acc2 = {};
#pragma unroll
        for (int k = 0; k < 5; ++k) {
            const int r0 = 4 * k + k0, r1 = 4 * k + k1;
            v2f b0, b1, b2;
            b0.x = P[r0 * PSTR + lcol];        b0.y = P[r1 * PSTR + lcol];
            b1.x = P[r0 * PSTR + 16 + lcol];   b1.y = P[r1 * PSTR + 16 + lcol];
            b2.x = P[r0 * PSTR + 32 + lcol];   b2.y = P[r1 * PSTR + 32 + lcol];
            acc0 = __builtin_amdgcn_wmma_f32_16x16x4_f32(false, wband[k], false, b0,
                                                         (short)0, acc0, false, false);
            acc1 = __builtin_amdgcn_wmma_f32_16x16x4_f32(false, wband[k], false, b1,
                                                         (short)0, acc1, false, false);
            acc2 = __builtin_amdgcn_wmma_f32_16x16x4_f32(false, wband[k], false, b2,
                                                         (short)0, acc2, false, false);
        }
#pragma unroll
        for (int i = 0; i < 8; ++i) {
            const int row = (i + 8 * hf) * TSTR;
            T[row + lcol]      = acc0[i];
            T[row + 16 + lcol] = acc1[i];
            T[row + 32 + lcol] = acc2[i];
        }
        __syncthreads();

        // ---- stage 2: Out tile t (cols 16t..16t+15) = T[:,16t..16t+19] band-sum ----
#pragma unroll
        for (int t = 0; t < 2; ++t) {
            v8f acc = {};
#pragma unroll
            for (int k = 0; k < 5; ++k) {
                v2f a;
                a.x = T[lcol * TSTR + 16 * t + 4 * k + k0];
                a.y = T[lcol * TSTR + 16 * t + 4 * k + k1];
                acc = __builtin_amdgcn_wmma_f32_16x16x4_f32(false, a, false, wband[k],
                                                            (short)0, acc, false, false);
            }
#pragma unroll
            for (int i = 0; i < 8; ++i) {
                const int   m   = i + 8 * hf;
                const float ctr = P[(m + 2) * PSTR + 16 * t + lcol + 2];
                sal[t][i] += __builtin_fabsf(ctr - acc[i] * 0.04f);   // /25
            }
        }
        __syncthreads();   // before next channel overwrites P
    }

    // ---- write unnormalized sal, wave-max, distributed atomicMax ----
    float mx = 0.f;
#pragma unroll
    for (int t = 0; t < 2; ++t)
#pragma unroll
        for (int i = 0; i < 8; ++i) {
            const int   m = i + 8 * hf;
            const float s = sal[t][i] * (1.f / 3.f);
            out[(size_t)b * (IMG_H * IMG_W) + (size_t)(y0 + m) * IMG_W
                + (x0 + 16 * t + lcol)] = s;
            mx = fmaxf(mx, s);
        }
#pragma unroll
    for (int off = 16; off > 0; off >>= 1)
        mx = fmaxf(mx, __shfl_xor(mx, off, 32));
    if (lane == 0) {
        const unsigned int slot =
            ((blockIdx.z << 11) + (blockIdx.y << 5) + blockIdx.x) & (NSLOT - 1);
        atomicMax(&slots[slot], __float_as_uint(mx));   // sal >= 0
    }
}

__global__ void init_ws_kernel(unsigned int* ws)
{
    const int i = blockIdx.x * blockDim.x + threadIdx.x;
    if (i <= NSLOT) ws[i] = 0u;      // slots[0..NSLOT-1] and final at [NSLOT]
}

__global__ void __launch_bounds__(256)
reduce_max_kernel(unsigned int* ws)
{
    __shared__ float sm[8];
    const int t = threadIdx.x;
    float m = 0.f;
    for (int i = t; i < NSLOT; i += 256)
        m = fmaxf(m, __uint_as_float(ws[i]));
#pragma unroll
    for (int off = 16; off > 0; off >>= 1)
        m = fmaxf(m, __shfl_xor(m, off, 32));
    if ((t & 31) == 0) sm[t >> 5] = m;
    __syncthreads();
    if (t < 8) {
        float v = sm[t];
#pragma unroll
        for (int off = 4; off > 0; off >>= 1)
            v = fmaxf(v, __shfl_xor(v, off, 32));
        if (t == 0) ws[NSLOT] = __float_as_uint(v);
    }
}

__global__ void __launch_bounds__(256)
norm_kernel(float4* __restrict__ out, const unsigned int* __restrict__ ws, int n4)
{
    const float mx = __uint_as_float(ws[NSLOT]);
    const float s  = 1.f / (mx + 1e-6f);
    const int  idx = blockIdx.x * blockDim.x + threadIdx.x;
    const int  str = gridDim.x * blockDim.x;
    for (int i = idx; i < n4; i += str) {
        float4 v = out[i];
        v.x *= s; v.y *= s; v.z *= s; v.w *= s;
        out[i] = v;
    }
}

extern "C" void kernel_launch(void* const* d_in, const int* in_sizes, int n_in,
                              void* d_out, int out_size, void* d_ws, size_t ws_size,
                              hipStream_t stream)
{
    (void)in_sizes; (void)n_in; (void)out_size; (void)ws_size;
    const float*  frame = (const float*)d_in[0];
    float*        out   = (float*)d_out;
    unsigned int* ws    = (unsigned int*)d_ws;

    hipLaunchKernelGGL(init_ws_kernel, dim3(5), dim3(256), 0, stream, ws);

    dim3 grid(IMG_W / 32, IMG_H / 16, NB);
    hipLaunchKernelGGL(sal5x5_wmma_kernel, grid, dim3(32), 0, stream,
                       frame, out, ws);

    hipLaunchKernelGGL(reduce_max_kernel, dim3(1), dim3(256), 0, stream, ws);

    const int n4 = (NB * IMG_H * IMG_W) / 4;
    hipLaunchKernelGGL(norm_kernel, dim3(4096), dim3(256), 0, stream,
                       (float4*)out, ws, n4);
}